// CoordsToNRF_24429773980211
// MI455X (gfx1250) — compile-verified
//
#include <hip/hip_runtime.h>
#include <cstdint>

// CoordsToNRF: out[b,p] = atom_nc[b,p] * (AU2KCALMOLA/MAX_NRF) / r2(b,p)
// Memory-bound: 33 MB in + 33 MB out + 1.5 MB coords @ 23.3 TB/s ~ 2.9 us.
//   * one block per batch element, 256 threads (8 wave32)
//   * coords[b] (1536 B) staged to LDS via gfx1250 async global->LDS b128
//     (ASYNCcnt path), overlapped with the first atom_nc b128 load (LOADcnt)
//   * atom_nc / out streamed as float4; software-pipelined next-tile prefetch
//   * pair (i,j): sqrt row-decode once per float4 group, then incremental
//     advance (j++ with row wrap) for the remaining 3 pairs

#define N_ATOMS 128
#define NC2     8128          // 128*127/2
#define NGROUPS (NC2 / 4)     // 2032 float4 groups per batch
#define THREADS 256

__global__ __launch_bounds__(THREADS)
void CoordsToNRF_kernel(const float* __restrict__ coords,   // [B,128,3]
                        const float* __restrict__ atom_nc,  // [B,8128]
                        float* __restrict__ out)            // [B,8128]
{
    constexpr float C = (float)((627.5095 * 0.529177) / 13036.0);

    __shared__ float sc[N_ATOMS * 3];   // raw 1536 B of coords[b]

    const int b   = blockIdx.x;
    const int tid = threadIdx.x;

    // ---- async stage coords[b] into LDS: 96 lanes x 16 B = 1536 B ----
    if (tid < (N_ATOMS * 3 * 4) / 16) {   // 96 lanes issue one b128 each
        // low 32 bits of a flat LDS pointer == LDS byte offset (aperture rule)
        uint32_t lds_off = (uint32_t)(uintptr_t)(const void*)&sc[tid * 4];
        const void* gsrc =
            (const void*)(coords + (size_t)b * (N_ATOMS * 3) + tid * 4);
        asm volatile("global_load_async_to_lds_b128 %0, %1, off"
                     :: "v"(lds_off), "v"(gsrc)
                     : "memory");
    }

    const float4* __restrict__ nc4  =
        (const float4*)(atom_nc + (size_t)b * NC2);
    float4* __restrict__ out4 = (float4*)(out + (size_t)b * NC2);

    // First stream tile: issue BEFORE waiting on the async copy so the HBM
    // stream overlaps the LDS staging (LOADcnt and ASYNCcnt are independent).
    int g = tid;                          // NGROUPS(2032) > THREADS(256)
    float4 cur = nc4[g];

    asm volatile("s_wait_asynccnt 0" ::: "memory");
    __syncthreads();

    while (true) {
        // prefetch next tile while computing the current one
        const int gn = g + THREADS;
        float4 nxt;
        if (gn < NGROUPS) nxt = nc4[gn];

        // ---- row decode once per group: i = floor((1+sqrt(8p+1))/2) ----
        const int p0 = g * 4;
        int i = (int)(0.5f * (1.0f + __builtin_sqrtf(8.0f * (float)p0 + 1.0f)));
        int ti = (i * (i - 1)) >> 1;        // triangular number T_{i-1}
        if (ti > p0)           { --i; ti -= i; }   // one-step fixup
        else if (ti + i <= p0) { ti += i; ++i; }
        int j = p0 - ti;

        float r2v[4];
        #pragma unroll
        for (int k = 0; k < 4; ++k) {
            const float dx = sc[3 * i + 0] - sc[3 * j + 0];
            const float dy = sc[3 * i + 1] - sc[3 * j + 1];
            const float dz = sc[3 * i + 2] - sc[3 * j + 2];
            r2v[k] = dx * dx + dy * dy + dz * dz;
            // advance to next flat pair: j++ with row wrap
            ++j;
            if (j == i) { j = 0; ++i; }
        }

        float4 o;
        o.x = cur.x * C / r2v[0];
        o.y = cur.y * C / r2v[1];
        o.z = cur.z * C / r2v[2];
        o.w = cur.w * C / r2v[3];
        out4[g] = o;                      // global_store_b128, coalesced

        if (gn >= NGROUPS) break;
        cur = nxt;
        g = gn;
    }
}

extern "C" void kernel_launch(void* const* d_in, const int* in_sizes, int n_in,
                              void* d_out, int out_size, void* d_ws, size_t ws_size,
                              hipStream_t stream) {
    const float* coords  = (const float*)d_in[0];   // [B,128,3] fp32
    const float* atom_nc = (const float*)d_in[1];   // [B,8128]  fp32
    float* out = (float*)d_out;                     // [B,8128]  fp32

    const int B = in_sizes[1] / NC2;                // 1024
    CoordsToNRF_kernel<<<B, THREADS, 0, stream>>>(coords, atom_nc, out);
}